// HeteroMLPPredictor_7739531067736
// MI455X (gfx1250) — compile-verified
//
#include <hip/hip_runtime.h>
#include <hip/hip_bf16.h>

// ---------------------------------------------------------------------------
// score[e] = h[src[e]] . W[0:128] + h[dst[e]] . W[128:256] + b
//
// Phase 1: per-node projections p[n] = (h[n].Wsrc, h[n].Wdst) via
//          V_WMMA_F32_16X16X4_F32 (full f32 precision, CDNA5 matrix pipe).
//          One wave32 per 16-node tile, K=128 consumed 4 at a time.
//          B-operand loads are branch-free (clamped address + data select)
//          so the 32 WMMAs pipeline without EXEC save/restore churn.
// Phase 2: per-edge gather-add from the 400KB p[] table (L2 resident);
//          index/output streams use non-temporal hints.
// ---------------------------------------------------------------------------

typedef __attribute__((ext_vector_type(2))) float v2f;
typedef __attribute__((ext_vector_type(8))) float v8f;
typedef __attribute__((ext_vector_type(4))) float v4f;
typedef __attribute__((ext_vector_type(4))) int   v4i;

#define IN_DIMS 128

__global__ void node_proj_wmma(const float* __restrict__ h,
                               const float* __restrict__ W,
                               float* __restrict__ p,   // 2 floats per node
                               int n_nodes) {
  const int lane = threadIdx.x & 31;
  const int wave = (blockIdx.x * blockDim.x + threadIdx.x) >> 5;
  const int node_base = wave * 16;              // 16 nodes per wave tile
  if (node_base >= n_nodes) return;             // wave-uniform guard

  const int half = lane >> 4;                   // 0: K={0,1}, 1: K={2,3}
  const int l    = lane & 15;
  const int kofs = half * 2;

  // A: lane l holds row (node_base+l), elements K=k+kofs, k+kofs+1
  const float* arow = h + (size_t)(node_base + l) * IN_DIMS + kofs;

  // B: column 0 = W_src, column 1 = W_dst, columns 2..15 = 0.
  // Always load from a valid address (lanes >=2 alias column 0), then
  // zero the data with a branch-free select -> no EXEC divergence.
  const bool live = (l < 2);
  const float* wbase = W + (live ? l * IN_DIMS : 0) + kofs;

  v8f c = {};
#pragma unroll
  for (int k = 0; k < IN_DIMS; k += 4) {
    v2f a  = *(const v2f*)(arow + k);           // 8B-aligned (even float idx)
    v2f wv = *(const v2f*)(wbase + k);
    v2f bv;
    bv.x = live ? wv.x : 0.0f;                  // v_cndmask, no branch
    bv.y = live ? wv.y : 0.0f;
    // D = A(16x4,f32) x B(4x16,f32) + C
    c = __builtin_amdgcn_wmma_f32_16x16x4_f32(
        /*neg_a=*/false, a, /*neg_b=*/false, bv,
        /*c_mod=*/(short)0, c, /*reuse_a=*/false, /*reuse_b=*/false);
  }

  // D layout: lane L<16, VGPR r -> (M=r, N=L); lane L+16, VGPR r -> (M=8+r, N=L)
  // Column 0 (lanes 0,16) = Wsrc proj; column 1 (lanes 1,17) = Wdst proj.
  if (live) {
    const int m0 = node_base + half * 8;
#pragma unroll
    for (int r = 0; r < 8; ++r) {
      p[2 * (m0 + r) + l] = c[r];
    }
  }
}

__global__ void edge_score(const int* __restrict__ src,
                           const int* __restrict__ dst,
                           const float* __restrict__ p,
                           const float* __restrict__ bptr,
                           float* __restrict__ out,
                           int n_edges) {
  const int e = (blockIdx.x * blockDim.x + threadIdx.x) * 4;
  if (e >= n_edges) return;
  const float bb = bptr[0];                     // scalar broadcast

  // Touch-once index streams: non-temporal so the p[] table stays cached.
  const v4i s = __builtin_nontemporal_load((const v4i*)(src + e));
  const v4i d = __builtin_nontemporal_load((const v4i*)(dst + e));

  v4f o;
  o.x = p[2 * s.x] + p[2 * d.x + 1] + bb;
  o.y = p[2 * s.y] + p[2 * d.y + 1] + bb;
  o.z = p[2 * s.z] + p[2 * d.z + 1] + bb;
  o.w = p[2 * s.w] + p[2 * d.w + 1] + bb;
  __builtin_nontemporal_store(o, (v4f*)(out + e));      // b128 NT store
}

extern "C" void kernel_launch(void* const* d_in, const int* in_sizes, int n_in,
                              void* d_out, int out_size, void* d_ws, size_t ws_size,
                              hipStream_t stream) {
  const float* h    = (const float*)d_in[0];
  const int*   src  = (const int*)d_in[1];
  const int*   dst  = (const int*)d_in[2];
  const float* W    = (const float*)d_in[3];
  const float* bptr = (const float*)d_in[4];
  float* out = (float*)d_out;
  float* p   = (float*)d_ws;                    // 2 floats per node (400 KB)

  const int n_nodes = in_sizes[0] / IN_DIMS;    // 50000
  const int n_edges = in_sizes[1];              // 800000

  // Phase 1: 16 nodes per wave, 8 waves (256 threads) per block
  const int tiles = (n_nodes + 15) / 16;
  const int waves_per_block = 256 / 32;
  const int blocks1 = (tiles + waves_per_block - 1) / waves_per_block;
  node_proj_wmma<<<blocks1, 256, 0, stream>>>(h, W, p, n_nodes);

  // Phase 2: 4 edges per thread
  const int threads2 = (n_edges + 3) / 4;
  const int blocks2 = (threads2 + 255) / 256;
  edge_score<<<blocks2, 256, 0, stream>>>(src, dst, p, bptr, out, n_edges);
}